// BatchTreeEncoder_84645215470007
// MI455X (gfx1250) — compile-verified
//
#include <hip/hip_runtime.h>
#include <hip/hip_bf16.h>
#include <cmath>

// ---------------------------------------------------------------------------
// BatchTreeEncoder on MI455X (gfx1250, wave32, WMMA).
//
// Child attention collapses to a per-node scalar gate:
//   chs[n] = h_last[n] * sigmoid(s_n - u0 - ln(S-1)),
//   s_n = tanh( tanh(h_last[n] @ sent_w + sent_b) . ctx ),
//   u0  = tanh( tanh(sent_b) . ctx )   (constant).
//
// GRU GEMMs (dominant: ~137 GFLOP f32, K=256, N=768) run on the native f32
// matrix path V_WMMA_F32_16X16X4_F32. Each workgroup (8 wave32) processes a
// 32-row tile: two 16-row M-subtiles share every B (weight) fragment, so each
// v2f weight load feeds two WMMAs (12 WMMA : 4 ds + 6 global loads per K-step).
// Activations are staged in LDS with a 260-float padded stride -> the
// per-K-step ds_load_b64 A-fragment reads are 64-bank conflict-free.
// ---------------------------------------------------------------------------

typedef float v2f __attribute__((ext_vector_type(2)));
typedef float v8f __attribute__((ext_vector_type(8)));

#define HDIM 256
#define LDSW 260   // padded LDS row stride (floats); 260 % 64 == 4 -> conflict-free

__device__ __forceinline__ float sigmoidf_(float x) { return 1.0f / (1.0f + __expf(-x)); }

__device__ __forceinline__ v8f wmma_f32(v2f a, v2f b, v8f c) {
  // D = A(16x4) * B(4x16) + C, all f32.
  // (neg_a, A, neg_b, B, c_mod, C, reuse_a, reuse_b)
  return __builtin_amdgcn_wmma_f32_16x16x4_f32(false, a, false, b, (short)0, c,
                                               false, false);
}

// ---------------------------------------------------------------------------
// Fused GRU level kernel: one workgroup per 32-row tile (two 16-row subtiles).
//   gi = X @ w_ih^T + b_ih ; gh = H0 @ w_hh^T + b_hh ; GRU gate math ; write h.
// Wave w owns output column tiles {w, w+8} (16 cols each); per tile it keeps
// 3 gate-column accumulators (r/z/n) x 2 matrices x 2 row-subtiles.
// ---------------------------------------------------------------------------
template <bool HASH>
__global__ __launch_bounds__(256) void gru_level_kernel(
    const int* __restrict__ tokens, const float* __restrict__ embedding,
    const float* __restrict__ w_ih, const float* __restrict__ w_hh,
    const float* __restrict__ b_ih, const float* __restrict__ b_hh,
    const float* __restrict__ h_init, float* __restrict__ h_out) {
  __shared__ float sX[32 * LDSW];
  __shared__ float sH[32 * LDSW];

  const int t = threadIdx.x;
  const int row0 = blockIdx.x * 32;

  // ---- stage embedding rows (gather) and H0 tile into LDS (float4) ----
  {
    const int r = t >> 3;          // 0..31
    const int cg = t & 7;          // 8 col groups of 32 floats
    const int tokenId = tokens[row0 + r];
    const float* erow = embedding + (size_t)tokenId * HDIM;
    const float* hrow = HASH ? (h_init + (size_t)(row0 + r) * HDIM) : nullptr;
#pragma unroll
    for (int q = 0; q < 8; ++q) {
      const int c = cg * 32 + q * 4;
      *(float4*)(&sX[r * LDSW + c]) = *(const float4*)(erow + c);
      if (HASH) *(float4*)(&sH[r * LDSW + c]) = *(const float4*)(hrow + c);
    }
  }
  __syncthreads();

  const int wave = t >> 5;
  const int lane = t & 31;
  const int half = lane >> 4;   // A/B K-half select per WMMA f32 layout
  const int m = lane & 15;

#pragma unroll
  for (int ot = 0; ot < 2; ++ot) {
    const int j0 = (wave + ot * 8) * 16;  // output hidden-column tile base
    // accumulators: [gate r/z/n][row subtile a/b]
    v8f giA[3] = {{0,0,0,0,0,0,0,0},{0,0,0,0,0,0,0,0},{0,0,0,0,0,0,0,0}};
    v8f giB[3] = {{0,0,0,0,0,0,0,0},{0,0,0,0,0,0,0,0},{0,0,0,0,0,0,0,0}};
    v8f ghA[3] = {{0,0,0,0,0,0,0,0},{0,0,0,0,0,0,0,0},{0,0,0,0,0,0,0,0}};
    v8f ghB[3] = {{0,0,0,0,0,0,0,0},{0,0,0,0,0,0,0,0},{0,0,0,0,0,0,0,0}};

    // B fragment base: w[(gatecol)*256 + k]; gate offsets 0 / 65536 / 131072
    const float* bi = w_ih + (size_t)(j0 + m) * HDIM + 2 * half;
    const float* bh = w_hh + (size_t)(j0 + m) * HDIM + 2 * half;

    for (int k0 = 0; k0 < HDIM; k0 += 4) {
      const int la = m * LDSW + k0 + 2 * half;         // row subtile a
      const int lb = (m + 16) * LDSW + k0 + 2 * half;  // row subtile b
      v2f axA = *(const v2f*)(&sX[la]);
      v2f axB = *(const v2f*)(&sX[lb]);
      v2f w0 = *(const v2f*)(bi + k0);
      v2f w1 = *(const v2f*)(bi + 65536 + k0);
      v2f w2 = *(const v2f*)(bi + 131072 + k0);
      giA[0] = wmma_f32(axA, w0, giA[0]);
      giB[0] = wmma_f32(axB, w0, giB[0]);
      giA[1] = wmma_f32(axA, w1, giA[1]);
      giB[1] = wmma_f32(axB, w1, giB[1]);
      giA[2] = wmma_f32(axA, w2, giA[2]);
      giB[2] = wmma_f32(axB, w2, giB[2]);
      if (HASH) {
        v2f ahA = *(const v2f*)(&sH[la]);
        v2f ahB = *(const v2f*)(&sH[lb]);
        v2f u0 = *(const v2f*)(bh + k0);
        v2f u1 = *(const v2f*)(bh + 65536 + k0);
        v2f u2 = *(const v2f*)(bh + 131072 + k0);
        ghA[0] = wmma_f32(ahA, u0, ghA[0]);
        ghB[0] = wmma_f32(ahB, u0, ghB[0]);
        ghA[1] = wmma_f32(ahA, u1, ghA[1]);
        ghB[1] = wmma_f32(ahB, u1, ghB[1]);
        ghA[2] = wmma_f32(ahA, u2, ghA[2]);
        ghB[2] = wmma_f32(ahB, u2, ghB[2]);
      }
    }

    // ---- GRU epilogue (D layout: vgpr v -> row v + 8*half, lane -> col) ----
    const int j = j0 + m;
    const float bir = b_ih[j], biz = b_ih[HDIM + j], bin = b_ih[2 * HDIM + j];
    const float bhr = b_hh[j], bhz = b_hh[HDIM + j], bhn = b_hh[2 * HDIM + j];
#pragma unroll
    for (int sub = 0; sub < 2; ++sub) {
      const v8f* gi = sub ? giB : giA;
      const v8f* gh = sub ? ghB : ghA;
      const int rbase = sub * 16;
#pragma unroll
      for (int v = 0; v < 8; ++v) {
        const int lr = rbase + v + 8 * half;
        float gr = gi[0][v] + bir;
        float gz = gi[1][v] + biz;
        float gn = gi[2][v] + bin;
        float hr = bhr, hz = bhz, hn = bhn, h0 = 0.0f;
        if (HASH) {
          hr += gh[0][v]; hz += gh[1][v]; hn += gh[2][v];
          h0 = sH[lr * LDSW + j];
        }
        const float rr = sigmoidf_(gr + hr);
        const float zz = sigmoidf_(gz + hz);
        const float nn = tanhf(gn + rr * hn);
        h_out[(size_t)(row0 + lr) * HDIM + j] = (1.0f - zz) * nn + zz * h0;
      }
    }
  }
}

// ---------------------------------------------------------------------------
// Child-attention gate kernel (collapsed form). One workgroup per 16 nodes of
// level d: load last-child rows, T1 = tanh(HL @ sent_w + b) via WMMA,
// s = tanh(T1 . ctx), chs = HL * sigmoid(s - u0 - ln(S-1)).
// ---------------------------------------------------------------------------
__global__ __launch_bounds__(256) void child_gate_kernel(
    const float* __restrict__ h_child,   // level d+1 rows (node n's last child = row 4n+3)
    const float* __restrict__ sent_w,    // (H,H) row-major
    const float* __restrict__ sent_b,    // (H)
    const float* __restrict__ ctx_w,     // (H)
    float* __restrict__ chs,             // (N,H) out
    float lnSm1) {
  __shared__ float sHL[16 * LDSW];
  __shared__ float sU[16];
  __shared__ float sGate[16];
  __shared__ float sU0;

  const int t = threadIdx.x;
  const int row0 = blockIdx.x * 16;
  const int r = t >> 4;
  const int cg = t & 15;

  {
    const float* src = h_child + ((size_t)(row0 + r) * 4 + 3) * HDIM;
#pragma unroll
    for (int q = 0; q < 4; ++q) {
      const int c = cg * 16 + q * 4;
      *(float4*)(&sHL[r * LDSW + c]) = *(const float4*)(src + c);
    }
  }
  if (t < 16) sU[t] = 0.0f;
  if (t == 16) sU0 = 0.0f;
  __syncthreads();

  // constant u0 inner sum: sum_k tanh(b_k)*ctx_k
  atomicAdd(&sU0, tanhf(sent_b[t]) * ctx_w[t]);

  const int wave = t >> 5;
  const int lane = t & 31;
  const int half = lane >> 4;
  const int m = lane & 15;

#pragma unroll
  for (int ot = 0; ot < 2; ++ot) {
    const int c0 = (wave * 2 + ot) * 16;
    const int c = c0 + m;
    v8f acc = {0,0,0,0,0,0,0,0};
    for (int k0 = 0; k0 < HDIM; k0 += 4) {
      v2f aa = *(const v2f*)(&sHL[m * LDSW + k0 + 2 * half]);
      const int kk = k0 + 2 * half;
      v2f bb;
      bb.x = sent_w[(size_t)kk * HDIM + c];
      bb.y = sent_w[(size_t)(kk + 1) * HDIM + c];
      acc = wmma_f32(aa, bb, acc);
    }
    const float cw = ctx_w[c];
    const float sb = sent_b[c];
#pragma unroll
    for (int v = 0; v < 8; ++v) {
      // row = v + 8*half for all 16 lanes of this half -> reduce across them
      float p = tanhf(acc[v] + sb) * cw;
      p += __shfl_xor(p, 8, 16);
      p += __shfl_xor(p, 4, 16);
      p += __shfl_xor(p, 2, 16);
      p += __shfl_xor(p, 1, 16);
      if (m == 0) atomicAdd(&sU[v + 8 * half], p);
    }
  }
  __syncthreads();
  if (t < 16) {
    const float u0v = tanhf(sU0);
    sGate[t] = sigmoidf_(tanhf(sU[t]) - u0v - lnSm1);
  }
  __syncthreads();
  {
    const float g = sGate[r];
    float* dst = chs + (size_t)(row0 + r) * HDIM;
#pragma unroll
    for (int q = 0; q < 4; ++q) {
      const int c = cg * 16 + q * 4;
      float4 vh = *(const float4*)(&sHL[r * LDSW + c]);
      vh.x *= g; vh.y *= g; vh.z *= g; vh.w *= g;
      *(float4*)(dst + c) = vh;
    }
  }
}

// ---------------------------------------------------------------------------
// Final reduction: out[b,j] = max(0, h0[b,j], last-sibling rows of tree b at
// every level). max(0, ...) subsumes the relu on h0.
// ---------------------------------------------------------------------------
__global__ __launch_bounds__(256) void final_max_kernel(
    const float* __restrict__ h_all, float* __restrict__ out) {
  const long long OFFS[7] = {0, 32, 160, 672, 2720, 10912, 43680};
  const int b = blockIdx.x;    // tree
  const int j = threadIdx.x;   // hidden index
  float mv = fmaxf(0.0f, h_all[(size_t)b * HDIM + j]);  // level 0, row b
  long long per = 1;
  for (int d = 1; d <= 6; ++d) {
    per *= 4;
    const long long base = OFFS[d] + (long long)b * per;
    for (long long i = 3; i < per; i += 4)  // last sibling: i % 4 == 3
      mv = fmaxf(mv, h_all[(size_t)(base + i) * HDIM + j]);
  }
  out[(size_t)b * HDIM + j] = mv;
}

// ---------------------------------------------------------------------------
extern "C" void kernel_launch(void* const* d_in, const int* in_sizes, int n_in,
                              void* d_out, int out_size, void* d_ws, size_t ws_size,
                              hipStream_t stream) {
  (void)in_sizes; (void)n_in; (void)out_size; (void)ws_size;
  const int*   tokens    = (const int*)d_in[0];
  const float* embedding = (const float*)d_in[1];
  const float* w_ih      = (const float*)d_in[2];
  const float* w_hh      = (const float*)d_in[3];
  const float* b_ih      = (const float*)d_in[4];
  const float* b_hh      = (const float*)d_in[5];
  const float* sent_w    = (const float*)d_in[6];
  const float* sent_b    = (const float*)d_in[7];
  const float* ctx_w     = (const float*)d_in[8];

  static const long long LVL[7]  = {32, 128, 512, 2048, 8192, 32768, 131072};
  static const long long OFFS[8] = {0, 32, 160, 672, 2720, 10912, 43680, 174752};

  float* h_all = (float*)d_ws;                        // 174752*256 f32 (~179 MB)
  float* chs   = h_all + (size_t)174752 * HDIM;       // up to 32768*256 f32 (~34 MB)

  // Level 6 (leaves): h0 = 0
  gru_level_kernel<false><<<dim3((unsigned)(LVL[6] / 32)), dim3(256), 0, stream>>>(
      tokens + OFFS[6], embedding, w_ih, w_hh, b_ih, b_hh, nullptr,
      h_all + (size_t)OFFS[6] * HDIM);

  for (int d = 5; d >= 0; --d) {
    const float S = (d == 0) ? 32.0f : 4.0f;
    const float lnSm1 = logf(S - 1.0f);
    child_gate_kernel<<<dim3((unsigned)(LVL[d] / 16)), dim3(256), 0, stream>>>(
        h_all + (size_t)OFFS[d + 1] * HDIM, sent_w, sent_b, ctx_w, chs, lnSm1);
    gru_level_kernel<true><<<dim3((unsigned)(LVL[d] / 32)), dim3(256), 0, stream>>>(
        tokens + OFFS[d], embedding, w_ih, w_hh, b_ih, b_hh, chs,
        h_all + (size_t)OFFS[d] * HDIM);
  }

  final_max_kernel<<<dim3(32), dim3(256), 0, stream>>>(h_all, (float*)d_out);
}